// CrimeModelLSTM_26783416058293
// MI455X (gfx1250) — compile-verified
//
#include <hip/hip_runtime.h>
#include <hip/hip_bf16.h>

// Fused 2-layer LSTM (H=8) + FC head for MI455X (gfx1250, wave32).
//
// Per wave (one 16-row batch tile):
//   G^T[32,16] = W[32,8] @ H^T[8,16]  via v_wmma_f32_16x16x4_f32:
//     - gate-half m-tiles (gates 0..15 = i|f, gates 16..31 = g|o)
//     - K=8 split into two K=4 chunks, accumulated through the C operand
//   D layout (ISA 7.12.2): VGPR r, lane l -> M = r + 8*(l>>4), N = l&15.
//   => lane l holds, for batch (l&15): lanes 0-15 get {i,g}, lanes 16-31 get
//      {f,o}. Batched ds_swizzle(SWAPX16) exchanges make ALL lanes hold the
//      full h/c state redundantly.
//   B (4x16) lane layout assumed to mirror documented A (16x4) layout:
//     VGPR j, lane l: N = l&15, K = j + 2*(l>>4).
//
// Loop rotation for ILP: iteration t computes {L2(t), L1(t+1)}, which are
// independent (both consume h1(t)); the h-states are loop-carried directly as
// WMMA B operands, so each iteration OPENS with 12 back-to-back WMMAs.
// Select strategy: cross-array half-lane selects (si/sf/tg/so) use ternaries
// (single v_cndmask, dual-issuable); same-array B-operand builds use a
// bitmask select (2 ops) to defeat clang's select-of-indices fold that would
// otherwise lower them as 8-way compare/cndmask extract trees.
// Activations: gfx1250 v_tanh_f32 (1 TRANS op); sigmoid(x)=0.5*tanh(x/2)+0.5;
// lane-half scaling folded into precomputed per-lane constants.

typedef float v2f __attribute__((ext_vector_type(2)));
typedef float v8f __attribute__((ext_vector_type(8)));

__device__ __forceinline__ v8f wmma4(v2f a, v2f b, v8f c) {
  // (neg_a, A, neg_b, B, c_mod, C, reuse_a, reuse_b)
  return __builtin_amdgcn_wmma_f32_16x16x4_f32(false, a, false, b, (short)0, c,
                                               false, false);
}

__device__ __forceinline__ float ftanh(float x) {
#if __has_builtin(__builtin_amdgcn_tanhf)
  return __builtin_amdgcn_tanhf(x);
#elif __has_builtin(__builtin_amdgcn_tanh_f32)
  return __builtin_amdgcn_tanh_f32(x);
#else
  float e = __builtin_amdgcn_exp2f(2.8853900817779268f * x);
  return __builtin_fmaf(-2.0f, __builtin_amdgcn_rcpf(1.0f + e), 1.0f);
#endif
}

__device__ __forceinline__ float fsig(float x) {
  return __builtin_fmaf(0.5f, ftanh(0.5f * x), 0.5f);
}

// Exchange with lane (l ^ 16): ds_swizzle group-of-32, xor=0x10, and=0x1f.
__device__ __forceinline__ float xor16(float v) {
  return __int_as_float(__builtin_amdgcn_ds_swizzle(__float_as_int(v), 0x401F));
}

__global__ __launch_bounds__(256) void lstm2_fused_kernel(
    const float* __restrict__ x,
    const float* __restrict__ W_ih1, const float* __restrict__ W_hh1,
    const float* __restrict__ b_ih1, const float* __restrict__ b_hh1,
    const float* __restrict__ W_ih2, const float* __restrict__ W_hh2,
    const float* __restrict__ b_ih2, const float* __restrict__ b_hh2,
    const float* __restrict__ W_fc, const float* __restrict__ b_fc,
    float* __restrict__ out, int B, int T) {
  const int lane = threadIdx.x & 31;
  const int wid = (int)((blockIdx.x * blockDim.x + threadIdx.x) >> 5);
  const int b0 = wid << 4;           // 16 batch rows per wave
  if (b0 >= B) return;               // wave-uniform: EXEC stays all-ones
  const int n = lane & 15;           // batch column within tile
  const int up = lane >> 4;          // 0 = lower half-lanes, 1 = upper

  // hi-gate activation = pA * tanh(sIn * pre) + pB:
  //   lower lanes: tanh(pre)   (g gate) -> sIn=1,  pA=1,  pB=0
  //   upper lanes: sigmoid(pre)(o gate) -> sIn=.5, pA=.5, pB=.5
  const float sIn = up ? 0.5f : 1.0f;
  const float pA  = up ? 0.5f : 1.0f;
  const float pB  = up ? 0.5f : 0.0f;

  // fold-proof half-lane select for SAME-ARRAY operands: up ? hi : lo
  const unsigned selM  = up ? 0xFFFFFFFFu : 0u;
  const unsigned selMn = ~selM;
  auto hsel = [&](float hiv, float lov) {
    return __uint_as_float((selM & __float_as_uint(hiv)) |
                           (selMn & __float_as_uint(lov)));
  };

  // ---- per-lane loop-invariant constants -------------------------------
  float bias1lo[8], bias1hi[8], wx1lo[8], wx1hi[8], bias2lo[8], bias2hi[8];
#pragma unroll
  for (int r = 0; r < 8; ++r) {
    int gl = r + 8 * up;
    int gh = gl + 16;
    bias1lo[r] = b_ih1[gl] + b_hh1[gl];
    bias1hi[r] = b_ih1[gh] + b_hh1[gh];
    wx1lo[r] = W_ih1[gl];  // W_ih1 is [32,1]
    wx1hi[r] = W_ih1[gh];
    bias2lo[r] = b_ih2[gl] + b_hh2[gl];
    bias2hi[r] = b_ih2[gh] + b_hh2[gh];
  }

  // A operand (16x4 f32 layout): lane l: M = l&15, VGPR j: K = j + 2*up.
  auto ldA = [&](const float* __restrict__ W, int half, int c) {
    int g = 16 * half + n;
    int k = 2 * up + 4 * c;
    v2f a;
    a.x = W[g * 8 + k];
    a.y = W[g * 8 + k + 1];
    return a;
  };
  v2f aW1[2][2], aI2[2][2], aW2[2][2];
#pragma unroll
  for (int h = 0; h < 2; ++h) {
#pragma unroll
    for (int c = 0; c < 2; ++c) {
      aW1[h][c] = ldA(W_hh1, h, c);
      aI2[h][c] = ldA(W_ih2, h, c);
      aW2[h][c] = ldA(W_hh2, h, c);
    }
  }

  // ---- state ------------------------------------------------------------
  // c1/c2: cell states (redundant in both halves). h-states are carried
  // directly as WMMA B operands (B-layout: VGPR j <- h[4*chunk + 2*up + j]);
  // h2 additionally kept as scalars for the FC epilogue.
  float c1[8], c2[8], h2[8];
#pragma unroll
  for (int r = 0; r < 8; ++r) { c1[r] = 0.0f; c2[r] = 0.0f; h2[r] = 0.0f; }
  v2f B1c0 = {0.0f, 0.0f}, B1c1 = {0.0f, 0.0f};
  v2f B2c0 = {0.0f, 0.0f}, B2c1 = {0.0f, 0.0f};

  // Fused gate update for both layers: all TRANS stream together, all 32
  // ds_swizzle exchanges share one dscnt wait. Produces next-step B operands
  // from fresh SSA scalars. With D2==0 this preserves zero layer-2 state
  // exactly (sig(0)=0.5, tanh(0)=0), which the preamble exploits.
  auto gates2 = [&](const v8f& D1lo, const v8f& D1hi, const v8f& D2lo,
                    const v8f& D2hi) {
    float s1lo[8], s1hi[8], s2lo[8], s2hi[8];
#pragma unroll
    for (int r = 0; r < 8; ++r) {
      s1lo[r] = fsig(D1lo[r]);                                 // sig(i)|sig(f)
      s1hi[r] = __builtin_fmaf(pA, ftanh(sIn * D1hi[r]), pB);  // tanh(g)|sig(o)
      s2lo[r] = fsig(D2lo[r]);
      s2hi[r] = __builtin_fmaf(pA, ftanh(sIn * D2hi[r]), pB);
    }
    float y1lo[8], y1hi[8], y2lo[8], y2hi[8];
#pragma unroll
    for (int r = 0; r < 8; ++r) {
      y1lo[r] = xor16(s1lo[r]);
      y1hi[r] = xor16(s1hi[r]);
      y2lo[r] = xor16(s2lo[r]);
      y2hi[r] = xor16(s2hi[r]);
    }
    float h1n[8], h2n[8];
#pragma unroll
    for (int r = 0; r < 8; ++r) {
      // cross-array selects -> single v_cndmask each (cannot fold to extract)
      float si1 = up ? y1lo[r] : s1lo[r];
      float sf1 = up ? s1lo[r] : y1lo[r];
      float tg1 = up ? y1hi[r] : s1hi[r];
      float so1 = up ? s1hi[r] : y1hi[r];
      float cc1 = __builtin_fmaf(sf1, c1[r], si1 * tg1);
      c1[r] = cc1;
      h1n[r] = so1 * ftanh(cc1);

      float si2 = up ? y2lo[r] : s2lo[r];
      float sf2 = up ? s2lo[r] : y2lo[r];
      float tg2 = up ? y2hi[r] : s2hi[r];
      float so2 = up ? s2hi[r] : y2hi[r];
      float cc2 = __builtin_fmaf(sf2, c2[r], si2 * tg2);
      c2[r] = cc2;
      float hh2 = so2 * ftanh(cc2);
      h2n[r] = hh2;
      h2[r] = hh2;  // scalar copy for the FC epilogue
    }
    // next-step B operands: same-array selects -> fold-proof bitmask select
    B1c0.x = hsel(h1n[2], h1n[0]);
    B1c0.y = hsel(h1n[3], h1n[1]);
    B1c1.x = hsel(h1n[6], h1n[4]);
    B1c1.y = hsel(h1n[7], h1n[5]);
    B2c0.x = hsel(h2n[2], h2n[0]);
    B2c0.y = hsel(h2n[3], h2n[1]);
    B2c1.x = hsel(h2n[6], h2n[4]);
    B2c1.y = hsel(h2n[7], h2n[5]);
  };

  const float* __restrict__ xrow = x + (size_t)(b0 + n) * (size_t)T;

  // ---- preamble: L1(0). h1=c1=0 => recurrent WMMA contribution is 0, so
  // D1 = C-init exactly; D2 = 0 keeps layer-2 state at exactly zero.
  float xt = xrow[0];
  float xn = xrow[(T > 1) ? 1 : 0];  // x[1] for L1(1) in iteration 0
  {
    v8f D1lo, D1hi;
    v8f Z = {};
#pragma unroll
    for (int r = 0; r < 8; ++r) {
      D1lo[r] = __builtin_fmaf(wx1lo[r], xt, bias1lo[r]);
      D1hi[r] = __builtin_fmaf(wx1hi[r], xt, bias1hi[r]);
    }
    gates2(D1lo, D1hi, Z, Z);  // -> B1 = h1(0); B2 stays 0
  }

  // ---- rotated main loop: iteration t does {L2(t), L1(t+1)} --------------
  for (int t = 0; t < T; ++t) {
    // prefetch x[t+2] one full iteration before its use (clamped, uniform)
    int t2 = (t + 2 < T) ? (t + 2) : (T - 1);
    float xn2 = xrow[t2];

    // L2(t) C-init (bias) and L1(t+1) C-init (bias + W_ih1 (x) x[t+1])
    v8f D2lo, D2hi, D1lo, D1hi;
#pragma unroll
    for (int r = 0; r < 8; ++r) {
      D2lo[r] = bias2lo[r];
      D2hi[r] = bias2hi[r];
      D1lo[r] = __builtin_fmaf(wx1lo[r], xn, bias1lo[r]);
      D1hi[r] = __builtin_fmaf(wx1hi[r], xn, bias1hi[r]);
    }

    // 12 WMMAs over two independent chains (L2: 8, L1: 4); B1 feeds both
    // L2's input projection and L1's recurrence, B2 feeds L2's recurrence.
    D2lo = wmma4(aI2[0][0], B1c0, D2lo);
    D2lo = wmma4(aI2[0][1], B1c1, D2lo);
    D1lo = wmma4(aW1[0][0], B1c0, D1lo);
    D1lo = wmma4(aW1[0][1], B1c1, D1lo);
    D2hi = wmma4(aI2[1][0], B1c0, D2hi);
    D2hi = wmma4(aI2[1][1], B1c1, D2hi);
    D1hi = wmma4(aW1[1][0], B1c0, D1hi);
    D1hi = wmma4(aW1[1][1], B1c1, D1hi);
    D2lo = wmma4(aW2[0][0], B2c0, D2lo);
    D2lo = wmma4(aW2[0][1], B2c1, D2lo);
    D2hi = wmma4(aW2[1][0], B2c0, D2hi);
    D2hi = wmma4(aW2[1][1], B2c1, D2hi);

    // both layers' nonlinearities fused (L1 result for step t+1 is dead work
    // on the final iteration; clamped x keeps it in-bounds and uniform)
    gates2(D1lo, D1hi, D2lo, D2hi);

    xt = xn;
    xn = xn2;
  }

  // =================== FC head: out[b,0:4] = h2 @ W_fc^T + b_fc ==========
  if (up == 0) {  // lanes 0-15 each own one batch row; no WMMA after here
    float* __restrict__ o = out + (size_t)(b0 + n) * 4;
#pragma unroll
    for (int j = 0; j < 4; ++j) {
      float acc = b_fc[j];
#pragma unroll
      for (int k = 0; k < 8; ++k) {
        acc = __builtin_fmaf(h2[k], W_fc[j * 8 + k], acc);
      }
      o[j] = acc;
    }
  }
}

extern "C" void kernel_launch(void* const* d_in, const int* in_sizes, int n_in,
                              void* d_out, int out_size, void* d_ws,
                              size_t ws_size, hipStream_t stream) {
  const float* x     = (const float*)d_in[0];
  const float* W_ih1 = (const float*)d_in[1];
  const float* W_hh1 = (const float*)d_in[2];
  const float* b_ih1 = (const float*)d_in[3];
  const float* b_hh1 = (const float*)d_in[4];
  const float* W_ih2 = (const float*)d_in[5];
  const float* W_hh2 = (const float*)d_in[6];
  const float* b_ih2 = (const float*)d_in[7];
  const float* b_hh2 = (const float*)d_in[8];
  const float* W_fc  = (const float*)d_in[9];
  const float* b_fc  = (const float*)d_in[10];

  const int B = out_size / 4;            // 4096
  const int T = in_sizes[0] / B;         // 512

  const int waves = (B + 15) / 16;       // 256 waves, one 16-row tile each
  const int threads = waves * 32;
  const int block = 256;
  const int grid = (threads + block - 1) / block;

  lstm2_fused_kernel<<<grid, block, 0, stream>>>(
      x, W_ih1, W_hh1, b_ih1, b_hh1, W_ih2, W_hh2, b_ih2, b_hh2, W_fc, b_fc,
      (float*)d_out, B, T);
}